// RoIHeads_90314572300711
// MI455X (gfx1250) — compile-verified
//
#include <hip/hip_runtime.h>
#include <cstdint>
#include <cstddef>

// ---------------- problem constants (match reference) ----------------
constexpr int   B_IMG   = 4;
constexpr int   N_PROP  = 8000;
constexpr int   C_CLS   = 91;          // incl. background class 0
constexpr float IMG_W   = 1333.0f;
constexpr float IMG_H   = 800.0f;
constexpr float WX = 10.0f, WY = 10.0f, WW = 5.0f, WH = 5.0f;
constexpr float BBOX_XFORM_CLIP = 4.135166556742356f;   // log(1000/16)
constexpr float SCORE_THRESH = 0.05f;
constexpr float NMS_THRESH   = 0.5f;
constexpr int   DET_PER_IMG  = 100;
constexpr float MIN_SIZE     = 0.01f;
constexpr float NEG          = -1e30f;

constexpr int TILE_ROWS = 128;                       // proposals per workgroup in phase A
constexpr int TILES     = (N_PROP + TILE_ROWS - 1) / TILE_ROWS;  // 63
constexpr int NMS_BDIM  = 1024;                      // 32 waves (wave32)

typedef unsigned int v4u __attribute__((ext_vector_type(4)));
typedef int          v4i __attribute__((ext_vector_type(4)));
typedef int          v8i __attribute__((ext_vector_type(8)));

// ---------------- phase 0: zero the per-image candidate counters ----------------
__global__ void roi_init_kernel(int* cnt) {
    if (threadIdx.x < B_IMG) cnt[threadIdx.x] = 0;
}

// ---------------- phase A: softmax + decode + threshold + compact ----------------
// One 128-thread block per 128 proposals of one image. Logits tile is staged into
// LDS with one Tensor Data Mover DMA (gfx1250 tensor_load_to_lds) per block.
__global__ __launch_bounds__(TILE_ROWS) void roi_decode_kernel(
        const float* __restrict__ logits,      // [B*N, C]
        const float* __restrict__ regression,  // [B*N, C*4]
        const float* __restrict__ proposals,   // [B, N, 4]
        int* __restrict__ cnt,                 // [B]
        float* __restrict__ pool,              // per-image SoA arrays
        int max_cand) {
    __shared__ __align__(16) float tile[TILE_ROWS * C_CLS];

    const int blk  = blockIdx.x;
    const int b    = blk / TILES;
    const int t    = blk % TILES;
    const int row0 = t * TILE_ROWS;
    const int rows = min(TILE_ROWS, N_PROP - row0);

    // ---- stage logits tile (rows x 91 f32, contiguous, 16B-aligned) into LDS ----
    const float*   gsrc    = logits + ((size_t)b * N_PROP + row0) * C_CLS;
    const int      totalF  = rows * C_CLS;       // <= 11648, multiple of 4
    const unsigned ldsbase = (unsigned)(uintptr_t)(&tile[0]);

#if __has_builtin(__builtin_amdgcn_tensor_load_to_lds) && __has_builtin(__builtin_amdgcn_s_wait_tensorcnt)
    if (threadIdx.x < 32) {   // one wave drives the TDM (EXEC is ignored by tensor ops)
        const unsigned long long ga = (unsigned long long)(uintptr_t)gsrc;
        // ---- D# group 0: count=1, lds_addr, 57-bit global_addr, type=2 ----
        v4u g0;
        g0.x = 1u;                                              // count=1, is_restore=0
        g0.y = ldsbase;                                         // lds_addr (bytes)
        g0.z = (unsigned)(ga & 0xFFFFFFFFu);                    // global_addr[31:0]
        g0.w = (unsigned)((ga >> 32) & 0x01FFFFFFu) | (2u << 30); // global_addr[56:32], type=2
        // ---- D# group 1: 1-D tile: data_size=4B, dim0=tile0=totalF, dim1=tile1=1 ----
        const unsigned uF = (unsigned)totalF;
        v8i g1;
        g1[0] = (int)(2u << 16);          // workgroup_mask=0, data_size=2 (4 bytes)
        g1[1] = (int)((uF & 0xFFFFu) << 16);          // tensor_dim0[15:0]
        g1[2] = (int)((uF >> 16) | (1u << 16));       // tensor_dim0[31:16], tensor_dim1=1
        g1[3] = (int)((uF & 0xFFFFu) << 16);          // tensor_dim1 hi=0, tile_dim0=totalF
        g1[4] = 1;                                    // tile_dim1=1, tile_dim2=0
        g1[5] = (int)uF;                              // tensor_dim0_stride[31:0]
        g1[6] = 0;                                    // stride hi, tensor_dim1_stride lo
        g1[7] = 0;                                    // tensor_dim1_stride hi
        v4i g2 = {0, 0, 0, 0};
        v4i g3 = {0, 0, 0, 0};
#if defined(__clang_major__) && (__clang_major__ >= 23)
        v8i gx = {0, 0, 0, 0, 0, 0, 0, 0};
        __builtin_amdgcn_tensor_load_to_lds(g0, g1, g2, g3, gx, 0);
#else
        __builtin_amdgcn_tensor_load_to_lds(g0, g1, g2, g3, 0);
#endif
        __builtin_amdgcn_s_wait_tensorcnt(0);
    }
#else
    // fallback: per-lane async global->LDS copies (also a gfx1250-only path)
    const int chunks = totalF >> 2;
    for (int i = (int)threadIdx.x; i < chunks; i += (int)blockDim.x) {
        unsigned           l = ldsbase + ((unsigned)i << 4);
        unsigned long long g = (unsigned long long)(uintptr_t)(gsrc + (i << 2));
        asm volatile("global_load_async_to_lds_b128 %0, %1, off"
                     :: "v"(l), "v"(g) : "memory");
    }
    for (int f = (chunks << 2) + (int)threadIdx.x; f < totalF; f += (int)blockDim.x) {
        unsigned           l = ldsbase + ((unsigned)f << 2);
        unsigned long long g = (unsigned long long)(uintptr_t)(gsrc + f);
        asm volatile("global_load_async_to_lds_b32 %0, %1, off"
                     :: "v"(l), "v"(g) : "memory");
    }
    asm volatile("s_wait_asynccnt 0" ::: "memory");
#endif
    __syncthreads();

    if ((int)threadIdx.x >= rows) return;

    const int r = row0 + (int)threadIdx.x;            // proposal index within image
    const float* lrow = &tile[threadIdx.x * C_CLS];   // stride 91 (odd) -> conflict-free

    // softmax (max + sum of exp)
    float m = lrow[0];
    #pragma unroll 7
    for (int c = 1; c < C_CLS; ++c) m = fmaxf(m, lrow[c]);
    float s = 0.0f;
    #pragma unroll 7
    for (int c = 0; c < C_CLS; ++c) s += expf(lrow[c] - m);
    const float inv_s = 1.0f / s;

    // proposal geometry
    const float4 p  = reinterpret_cast<const float4*>(proposals)[(size_t)b * N_PROP + r];
    const float  pw = p.z - p.x;
    const float  ph = p.w - p.y;
    const float  cx = p.x + 0.5f * pw;
    const float  cy = p.y + 0.5f * ph;

    const float4* relrow =
        reinterpret_cast<const float4*>(regression + ((size_t)b * N_PROP + r) * (C_CLS * 4));
    __builtin_prefetch(relrow, 0, 1);   // global_prefetch_b8

    // per-image SoA candidate arrays
    const size_t S   = (size_t)max_cand;
    float* sc_a  = pool + (size_t)b * 7 * S;
    int*   gid_a = (int*)(sc_a + S);
    int*   lab_a = (int*)(sc_a + 2 * S);
    float* x1_a  = sc_a + 3 * S;
    float* y1_a  = sc_a + 4 * S;
    float* x2_a  = sc_a + 5 * S;
    float* y2_a  = sc_a + 6 * S;

    for (int c = 1; c < C_CLS; ++c) {
        const float score = expf(lrow[c] - m) * inv_s;
        if (score > SCORE_THRESH) {
            const float4 d  = relrow[c];
            const float  dx = d.x / WX;
            const float  dy = d.y / WY;
            const float  dw = fminf(d.z / WW, BBOX_XFORM_CLIP);
            const float  dh = fminf(d.w / WH, BBOX_XFORM_CLIP);
            const float  pcx = dx * pw + cx;
            const float  pcy = dy * ph + cy;
            const float  hw  = 0.5f * expf(dw) * pw;
            const float  hh  = 0.5f * expf(dh) * ph;
            float x1 = fminf(fmaxf(pcx - hw, 0.0f), IMG_W);
            float y1 = fminf(fmaxf(pcy - hh, 0.0f), IMG_H);
            float x2 = fminf(fmaxf(pcx + hw, 0.0f), IMG_W);
            float y2 = fminf(fmaxf(pcy + hh, 0.0f), IMG_H);
            if ((x2 - x1) >= MIN_SIZE && (y2 - y1) >= MIN_SIZE) {
                int idx = atomicAdd(&cnt[b], 1);
                if (idx < max_cand) {
                    sc_a[idx]  = score;
                    gid_a[idx] = r * (C_CLS - 1) + (c - 1);  // original flat order -> tie-break
                    lab_a[idx] = c;
                    x1_a[idx] = x1; y1_a[idx] = y1; x2_a[idx] = x2; y2_a[idx] = y2;
                }
            }
        }
    }
}

// ---------------- phase B: sequential NMS, one workgroup per image ----------------
__global__ __launch_bounds__(NMS_BDIM) void roi_nms_kernel(
        const int* __restrict__ cnt,
        float* __restrict__ pool,
        int max_cand,
        float* __restrict__ out) {      // boxes[1600] ++ scores[400] ++ labels[400]
    __shared__ float s_sc[NMS_BDIM / 32];
    __shared__ int   s_g [NMS_BDIM / 32];
    __shared__ int   s_i [NMS_BDIM / 32];
    __shared__ float w_x1, w_y1, w_x2, w_y2, w_area;
    __shared__ int   w_lab, w_ok;

    const int b    = blockIdx.x;
    const int tid  = (int)threadIdx.x;
    const int lane = tid & 31;
    const int wid  = tid >> 5;

    const size_t S = (size_t)max_cand;
    float* sc_a  = pool + (size_t)b * 7 * S;
    int*   gid_a = (int*)(sc_a + S);
    int*   lab_a = (int*)(sc_a + 2 * S);
    float* x1_a  = sc_a + 3 * S;
    float* y1_a  = sc_a + 4 * S;
    float* x2_a  = sc_a + 5 * S;
    float* y2_a  = sc_a + 6 * S;

    const int M = min(cnt[b], max_cand);

    constexpr int SC_BASE = B_IMG * DET_PER_IMG * 4;                 // 1600
    constexpr int LB_BASE = SC_BASE + B_IMG * DET_PER_IMG;           // 2000

    for (int it = 0; it < DET_PER_IMG; ++it) {
        // ---- strided local argmax (tie: smaller original gid, as jnp.argmax) ----
        float bs = NEG;
        int   bg = 0x7fffffff;
        int   bi = -1;
        for (int i = tid; i < M; i += NMS_BDIM) {
            const float v = sc_a[i];
            const int   g = gid_a[i];
            if (v > bs || (v == bs && g < bg)) { bs = v; bg = g; bi = i; }
        }
        // ---- wave32 butterfly reduce ----
        #pragma unroll
        for (int off = 16; off > 0; off >>= 1) {
            const float ov = __shfl_xor(bs, off, 32);
            const int   og = __shfl_xor(bg, off, 32);
            const int   oi = __shfl_xor(bi, off, 32);
            if (ov > bs || (ov == bs && og < bg)) { bs = ov; bg = og; bi = oi; }
        }
        if (lane == 0) { s_sc[wid] = bs; s_g[wid] = bg; s_i[wid] = bi; }
        __syncthreads();

        // ---- final reduce across the 32 waves, done by wave 0 ----
        if (wid == 0) {
            bs = s_sc[lane]; bg = s_g[lane]; bi = s_i[lane];
            #pragma unroll
            for (int off = 16; off > 0; off >>= 1) {
                const float ov = __shfl_xor(bs, off, 32);
                const int   og = __shfl_xor(bg, off, 32);
                const int   oi = __shfl_xor(bi, off, 32);
                if (ov > bs || (ov == bs && og < bg)) { bs = ov; bg = og; bi = oi; }
            }
            if (lane == 0) {
                const bool ok = (bi >= 0) && (bs > 0.5f * NEG);
                const int  o4 = (b * DET_PER_IMG + it) * 4;
                const int  o1 = b * DET_PER_IMG + it;
                if (ok) {
                    const float x1 = x1_a[bi], y1 = y1_a[bi], x2 = x2_a[bi], y2 = y2_a[bi];
                    out[o4 + 0] = x1; out[o4 + 1] = y1; out[o4 + 2] = x2; out[o4 + 3] = y2;
                    out[SC_BASE + o1] = bs;
                    out[LB_BASE + o1] = (float)lab_a[bi];
                    w_x1 = x1; w_y1 = y1; w_x2 = x2; w_y2 = y2;
                    w_area = (x2 - x1) * (y2 - y1);
                    w_lab  = lab_a[bi];
                    w_ok   = 1;
                } else {
                    out[o4 + 0] = 0.0f; out[o4 + 1] = 0.0f; out[o4 + 2] = 0.0f; out[o4 + 3] = 0.0f;
                    out[SC_BASE + o1] = 0.0f;
                    out[LB_BASE + o1] = 0.0f;
                    w_ok = 0;
                }
            }
        }
        __syncthreads();

        // ---- suppress same-class candidates with IoU > thresh (includes winner) ----
        if (w_ok) {
            const float wx1 = w_x1, wy1 = w_y1, wx2 = w_x2, wy2 = w_y2, wa = w_area;
            const int   wl  = w_lab;
            for (int i = tid; i < M; i += NMS_BDIM) {
                if (lab_a[i] == wl && sc_a[i] > 0.5f * NEG) {
                    const float ix1 = fmaxf(wx1, x1_a[i]);
                    const float iy1 = fmaxf(wy1, y1_a[i]);
                    const float ix2 = fminf(wx2, x2_a[i]);
                    const float iy2 = fminf(wy2, y2_a[i]);
                    const float iw = fmaxf(ix2 - ix1, 0.0f);
                    const float ih = fmaxf(iy2 - iy1, 0.0f);
                    const float inter = iw * ih;
                    const float a2 = (x2_a[i] - x1_a[i]) * (y2_a[i] - y1_a[i]);
                    const float iou = inter / (wa + a2 - inter);
                    if (iou > NMS_THRESH) sc_a[i] = NEG;
                }
            }
        }
        // no barrier needed here: next iteration's shared writes don't alias
        // the winner broadcast vars before the next __syncthreads()
    }
}

// ---------------- host entry ----------------
extern "C" void kernel_launch(void* const* d_in, const int* in_sizes, int n_in,
                              void* d_out, int out_size, void* d_ws, size_t ws_size,
                              hipStream_t stream) {
    (void)in_sizes; (void)n_in; (void)out_size;
    const float* logits     = (const float*)d_in[0];   // [B*N, C]
    const float* regression = (const float*)d_in[1];   // [B*N, C*4]
    const float* proposals  = (const float*)d_in[2];   // [B, N, 4]

    int* cnt    = (int*)d_ws;
    float* pool = (float*)((char*)d_ws + 256);

    // size candidate capacity from workspace (7 SoA f32 arrays per image)
    long mc = 1024;
    if (ws_size > 4096) {
        mc = (long)((ws_size - 256) / ((size_t)B_IMG * 7 * sizeof(float)));
        if (mc > (1L << 20)) mc = 1L << 20;
        if (mc < 1) mc = 1;
    }
    const int max_cand = (int)mc;

    roi_init_kernel<<<1, 64, 0, stream>>>(cnt);
    roi_decode_kernel<<<B_IMG * TILES, TILE_ROWS, 0, stream>>>(
        logits, regression, proposals, cnt, pool, max_cand);
    roi_nms_kernel<<<B_IMG, NMS_BDIM, 0, stream>>>(cnt, pool, max_cand, (float*)d_out);
}